// GNNModel_12000138625378
// MI455X (gfx1250) — compile-verified
//
#include <hip/hip_runtime.h>
#include <cstdint>
#include <cstddef>

#define N_NODES 20000
#define E_EDGES 320000
#define ET (E_EDGES + N_NODES)   // edges + self-loops = 340000
#define C_IN  256
#define C_HID 256
#define C_OUT 128
#define NEG_SLOPE 0.2f

typedef __attribute__((ext_vector_type(2))) float v2f;
typedef __attribute__((ext_vector_type(8))) float v8f;

// ---------------------------------------------------------------------------
// WMMA GEMM: C[M,F] = A[M,K] @ B[K,F], fp32, V_WMMA_F32_16X16X4_F32.
// One wave -> 16 rows x 64 cols (4 accumulators), K stepped by 4.
// A 16x4 layout: lanes 0-15 hold K={k0,k0+1}, lanes 16-31 hold K={k0+2,k0+3},
//                M = lane%16 (ISA 7.12.2, 32-bit A 16x4).
// B 4x16 layout: half-wave holds a K pair, N = lane%16 striped across lanes.
// C/D: VGPR r -> M = r + 8*(lane/16), N = lane%16.
// ---------------------------------------------------------------------------
template<int K, int F>
__global__ __launch_bounds__(256) void gemm_wmma_f32(
    const float* __restrict__ A, const float* __restrict__ B,
    float* __restrict__ C, int M) {
  const int wave = (int)((blockIdx.x * blockDim.x + threadIdx.x) >> 5);
  const int lane = threadIdx.x & 31;
  constexpr int colGroups = F / 64;
  const int rowTile  = wave / colGroups;
  const int colGroup = wave % colGroups;
  if (rowTile * 16 >= M) return;          // wave-uniform exit (EXEC stays all-1s)

  const int row0 = rowTile * 16;
  const int n0   = colGroup * 64;
  const int half = lane >> 4;             // 0: K pair lo, 1: K pair hi
  const int m    = lane & 15;

  v8f acc0 = {}, acc1 = {}, acc2 = {}, acc3 = {};
  const float* __restrict__ arow = A + (size_t)(row0 + m) * K;

  for (int k0 = 0; k0 < K; k0 += 4) {
    v2f a;
    a[0] = arow[k0 + 2 * half];
    a[1] = arow[k0 + 2 * half + 1];

    const float* __restrict__ bb = B + (size_t)(k0 + 2 * half) * F + n0 + m;
    v2f b0, b1, b2, b3;
    b0[0] = bb[0];  b0[1] = bb[F];
    b1[0] = bb[16]; b1[1] = bb[F + 16];
    b2[0] = bb[32]; b2[1] = bb[F + 32];
    b3[0] = bb[48]; b3[1] = bb[F + 48];

    acc0 = __builtin_amdgcn_wmma_f32_16x16x4_f32(false, a, false, b0, (short)0, acc0, false, false);
    acc1 = __builtin_amdgcn_wmma_f32_16x16x4_f32(false, a, false, b1, (short)0, acc1, false, false);
    acc2 = __builtin_amdgcn_wmma_f32_16x16x4_f32(false, a, false, b2, (short)0, acc2, false, false);
    acc3 = __builtin_amdgcn_wmma_f32_16x16x4_f32(false, a, false, b3, (short)0, acc3, false, false);
  }

  #pragma unroll
  for (int r = 0; r < 8; ++r) {
    const size_t row = (size_t)(row0 + r + 8 * half);
    float* __restrict__ crow = C + row * F + n0 + m;
    crow[0]  = acc0[r];
    crow[16] = acc1[r];
    crow[32] = acc2[r];
    crow[48] = acc3[r];
  }
}

// ---------------------------------------------------------------------------
// alpha_src[i] = dot(h[i,:], a_src); alpha_dst[i] = dot(h[i,:], a_dst).
// One wave per node, wave32 shuffle reduction.
// ---------------------------------------------------------------------------
template<int F>
__global__ __launch_bounds__(256) void alpha_kernel(
    const float* __restrict__ h, const float* __restrict__ a_src,
    const float* __restrict__ a_dst, float* __restrict__ as,
    float* __restrict__ ad, int n) {
  const int wave = (int)((blockIdx.x * blockDim.x + threadIdx.x) >> 5);
  const int lane = threadIdx.x & 31;
  if (wave >= n) return;
  const float* __restrict__ row = h + (size_t)wave * F;
  float s0 = 0.f, s1 = 0.f;
  #pragma unroll
  for (int j = 0; j < F / 32; ++j) {
    const int f = lane + 32 * j;
    const float v = row[f];
    s0 += v * a_src[f];
    s1 += v * a_dst[f];
  }
  #pragma unroll
  for (int off = 16; off > 0; off >>= 1) {
    s0 += __shfl_xor(s0, off, 32);
    s1 += __shfl_xor(s1, off, 32);
  }
  if (lane == 0) { as[wave] = s0; ad[wave] = s1; }
}

// ---------------------------------------------------------------------------
// Edge helpers. Edge e < E comes from edge_index (int64, [2,E] row-major);
// e >= E is the self-loop for node (e - E).
// ---------------------------------------------------------------------------
__device__ inline void edge_sd(const long long* __restrict__ ei, int e,
                               int& s, int& d) {
  if (e < E_EDGES) { s = (int)ei[e]; d = (int)ei[E_EDGES + e]; }
  else             { s = d = e - E_EDGES; }
}

// Order-preserving float atomic max (m initialized to -inf).
__device__ inline void atomicMaxF(float* addr, float v) {
  if (v >= 0.f) atomicMax((int*)addr, __float_as_int(v));
  else          atomicMin((unsigned int*)addr, __float_as_uint(v));
}

__global__ __launch_bounds__(256) void edge_pass1(
    const long long* __restrict__ ei, const float* __restrict__ as,
    const float* __restrict__ ad, float* __restrict__ ebuf,
    float* __restrict__ mx) {
  const int e = blockIdx.x * blockDim.x + threadIdx.x;
  if (e >= ET) return;
  int s, d; edge_sd(ei, e, s, d);
  float v = as[s] + ad[d];
  v = v > 0.f ? v : NEG_SLOPE * v;            // leaky_relu
  ebuf[e] = v;
  atomicMaxF(&mx[d], v);
}

__global__ __launch_bounds__(256) void edge_pass2(
    const long long* __restrict__ ei, float* __restrict__ ebuf,
    const float* __restrict__ mx, float* __restrict__ denom) {
  const int e = blockIdx.x * blockDim.x + threadIdx.x;
  if (e >= ET) return;
  int s, d; edge_sd(ei, e, s, d);
  const float w = __expf(ebuf[e] - mx[d]);
  ebuf[e] = w;
  atomicAdd(&denom[d], w);
}

// One block per edge, one thread per feature: fully coalesced gather of
// h[src,:] and scatter-add into agg[dst,:].
template<int F>
__global__ __launch_bounds__(F) void edge_aggregate(
    const long long* __restrict__ ei, const float* __restrict__ ebuf,
    const float* __restrict__ h, float* __restrict__ agg) {
  const int e = blockIdx.x;
  const int f = threadIdx.x;
  int s, d; edge_sd(ei, e, s, d);
  const float w = ebuf[e];
  atomicAdd(&agg[(size_t)d * F + f], w * h[(size_t)s * F + f]);
}

template<int F, bool RELU>
__global__ __launch_bounds__(256) void finalize_kernel(
    const float* __restrict__ agg, const float* __restrict__ denom,
    const float* __restrict__ bias, float* __restrict__ out, int n) {
  const size_t idx = (size_t)blockIdx.x * blockDim.x + threadIdx.x;
  if (idx >= (size_t)n * F) return;
  const int i = (int)(idx / F);
  const int f = (int)(idx % F);
  float v = agg[idx] / denom[i] + bias[f];
  if (RELU) v = v > 0.f ? v : 0.f;
  out[idx] = v;
}

__global__ __launch_bounds__(256) void init_node(float* __restrict__ mx,
                                                 float* __restrict__ denom,
                                                 int n) {
  const int i = blockIdx.x * blockDim.x + threadIdx.x;
  if (i < n) { mx[i] = -__builtin_inff(); denom[i] = 0.f; }
}

__global__ __launch_bounds__(256) void zero_fill(float* __restrict__ p, size_t n) {
  const size_t i = (size_t)blockIdx.x * blockDim.x + threadIdx.x;
  if (i < n) p[i] = 0.f;
}

// ---------------------------------------------------------------------------
extern "C" void kernel_launch(void* const* d_in, const int* in_sizes, int n_in,
                              void* d_out, int out_size, void* d_ws, size_t ws_size,
                              hipStream_t stream) {
  const float*     x   = (const float*)d_in[0];
  const long long* ei  = (const long long*)d_in[1];
  const float*     W1  = (const float*)d_in[2];
  const float*     as1 = (const float*)d_in[3];
  const float*     ad1 = (const float*)d_in[4];
  const float*     b1  = (const float*)d_in[5];
  const float*     W2  = (const float*)d_in[6];
  const float*     as2 = (const float*)d_in[7];
  const float*     ad2 = (const float*)d_in[8];
  const float*     b2  = (const float*)d_in[9];
  float* out = (float*)d_out;

  // Workspace layout (floats). h2 reuses h1's region (h1 dead after layer-1 agg).
  float* ws   = (float*)d_ws;
  float* h1   = ws;                                   // N * 256
  float* agg1 = h1   + (size_t)N_NODES * C_HID;       // N * 256 (-> hidden, in place)
  float* h2   = h1;                                   // N * 128 (alias h1)
  float* asrc = agg1 + (size_t)N_NODES * C_HID;       // N
  float* adst = asrc + N_NODES;                       // N
  float* mx   = adst + N_NODES;                       // N
  float* den  = mx   + N_NODES;                       // N
  float* ebuf = den  + N_NODES;                       // ET

  const int nodeBlk   = (N_NODES + 255) / 256;              // 79
  const int edgeBlk   = (ET + 255) / 256;                   // 1329
  const int alphaBlk  = (N_NODES * 32 + 255) / 256;         // 2500
  const int gemm1Blk  = (1250 * (C_HID / 64) * 32 + 255) / 256;   // 625
  const int gemm2Blk  = (1250 * (C_OUT / 64) * 32 + 255) / 256;   // 313
  const size_t n1     = (size_t)N_NODES * C_HID;            // 5.12M
  const size_t n2     = (size_t)N_NODES * C_OUT;            // 2.56M

  // ----------------- Layer 1 -----------------
  init_node<<<nodeBlk, 256, 0, stream>>>(mx, den, N_NODES);
  zero_fill<<<(int)((n1 + 255) / 256), 256, 0, stream>>>(agg1, n1);

  gemm_wmma_f32<C_IN, C_HID><<<gemm1Blk, 256, 0, stream>>>(x, W1, h1, N_NODES);
  alpha_kernel<C_HID><<<alphaBlk, 256, 0, stream>>>(h1, as1, ad1, asrc, adst, N_NODES);

  edge_pass1<<<edgeBlk, 256, 0, stream>>>(ei, asrc, adst, ebuf, mx);
  edge_pass2<<<edgeBlk, 256, 0, stream>>>(ei, ebuf, mx, den);
  edge_aggregate<C_HID><<<ET, C_HID, 0, stream>>>(ei, ebuf, h1, agg1);
  finalize_kernel<C_HID, true><<<(int)((n1 + 255) / 256), 256, 0, stream>>>(
      agg1, den, b1, agg1, N_NODES);   // hidden = relu(agg1/den + b1), in place

  // ----------------- Layer 2 -----------------
  init_node<<<nodeBlk, 256, 0, stream>>>(mx, den, N_NODES);
  zero_fill<<<(int)((n2 + 255) / 256), 256, 0, stream>>>(out, n2);

  gemm_wmma_f32<C_HID, C_OUT><<<gemm2Blk, 256, 0, stream>>>(agg1, W2, h2, N_NODES);
  alpha_kernel<C_OUT><<<alphaBlk, 256, 0, stream>>>(h2, as2, ad2, asrc, adst, N_NODES);

  edge_pass1<<<edgeBlk, 256, 0, stream>>>(ei, asrc, adst, ebuf, mx);
  edge_pass2<<<edgeBlk, 256, 0, stream>>>(ei, ebuf, mx, den);
  edge_aggregate<C_OUT><<<ET, C_OUT, 0, stream>>>(ei, ebuf, h2, out);
  finalize_kernel<C_OUT, false><<<(int)((n2 + 255) / 256), 256, 0, stream>>>(
      out, den, b2, out, N_NODES);     // out = agg2/den + b2, in place
}